// LossCorefLinkerESM_24790551232645
// MI455X (gfx1250) — compile-verified
//
#include <hip/hip_runtime.h>
#include <stdint.h>

// LossCorefLinkerESM for MI455X (gfx1250, wave32).
// loss[b,m] = LSE_unmasked(scores_row) - LSE_targets(scores_row), summed over
// rows with m < lengths[b].  Memory-bound: 34.6MB of scores @ 23.3TB/s ~1.5us.
// Rows are streamed with CDNA5 async global->LDS copies, triple-buffered via
// ASYNCcnt (2 rows in flight per wave; 2048 waves -> ~17MB outstanding).

#define B_   8
#define M_   1024
#define C_   32
#define N_   1056                 // C_ + M_
#define BLOCKS_  512
#define THREADS_ 128              // 4 waves
#define WAVES_PER_BLK 4
#define NWAVES_ (BLOCKS_ * WAVES_PER_BLK)   // 2048
#define ROWS_PER_WAVE ((B_ * M_) / NWAVES_) // 4
#define NBUF_ 3

typedef __attribute__((ext_vector_type(2))) float v2f;
typedef __attribute__((ext_vector_type(8))) float v8f;

__device__ __forceinline__ uint32_t lds_offset(const void* p) {
  // generic (flat) LDS pointer carries the byte offset in its low 32 bits
  return (uint32_t)(uint64_t)(uintptr_t)p;
}

__device__ __forceinline__ void async_b128(uint32_t lds_off, const void* g) {
  asm volatile("global_load_async_to_lds_b128 %0, %1, off"
               :: "v"(lds_off), "v"(g) : "memory");
}
__device__ __forceinline__ void async_b32(uint32_t lds_off, const void* g) {
  asm volatile("global_load_async_to_lds_b32 %0, %1, off"
               :: "v"(lds_off), "v"(g) : "memory");
}

__global__ __launch_bounds__(THREADS_) void coref_linker_loss_kernel(
    const float* __restrict__ scores,        // [B, M, N]
    const int*   __restrict__ link_targets,  // [B, M, C]
    const int*   __restrict__ cand_len,      // [B, M]
    const int*   __restrict__ cluster_ids,   // [B, M]
    const int*   __restrict__ lengths,       // [B]
    float*       __restrict__ partial)       // [BLOCKS_]
{
  __shared__ float smem[WAVES_PER_BLK * NBUF_ * N_];  // 50,688 B
  __shared__ float wacc[WAVES_PER_BLK];

  const int lane  = threadIdx.x & 31;
  const int wib   = threadIdx.x >> 5;
  const int gwave = blockIdx.x * WAVES_PER_BLK + wib;   // 0..2047
  const int m     = gwave & (M_ - 1);

  float* bufs[NBUF_];
  uint32_t lofs[NBUF_];
#pragma unroll
  for (int i = 0; i < NBUF_; ++i) {
    bufs[i] = &smem[(wib * NBUF_ + i) * N_];
    lofs[i] = lds_offset(bufs[i]);
  }

  // Issue one row (4224B) as 8x b128 + 1x b32 async copies (ASYNCcnt += 9)
  auto issue = [&](int row, uint32_t dstBase) {
    const char* src = (const char*)(scores + (size_t)row * N_);
    const uint32_t d = dstBase + (uint32_t)lane * 16u;
    const char* s = src + lane * 16;
#pragma unroll
    for (int i = 0; i < 8; ++i)
      async_b128(d + (uint32_t)i * 512u, s + i * 512);
    async_b32(dstBase + 4096u + (uint32_t)lane * 4u, src + 4096 + lane * 4);
  };

  float acc = 0.0f;
  issue(gwave, lofs[0]);
  issue(gwave + NWAVES_, lofs[1]);

#pragma unroll
  for (int k = 0; k < ROWS_PER_WAVE; ++k) {
    const float* rb = bufs[k % NBUF_];
    if (k + 2 < ROWS_PER_WAVE) {
      issue(gwave + (k + 2) * NWAVES_, lofs[(k + 2) % NBUF_]);
      // 3 rows (27 copies) may be outstanding; in-order completion => <=18
      // means the oldest row's 9 copies have landed in LDS.
      asm volatile("s_wait_asynccnt 18" ::: "memory");
    } else if (k + 1 < ROWS_PER_WAVE) {
      asm volatile("s_wait_asynccnt 9" ::: "memory");
    } else {
      asm volatile("s_wait_asynccnt 0" ::: "memory");
    }

    const int row = gwave + k * NWAVES_;
    const int b   = row >> 10;                         // row / M_
    const int lenb = lengths[b];
    if (m < lenb) {                                    // wave-uniform branch
      const int rowbase = b * M_ + m;
      const int clen = cand_len[rowbase];
      const int cidm = cluster_ids[rowbase];

      // ---- pass A: row max over unmasked columns ----
      float mx = -3.0e38f;
      {
        const float s0 = rb[lane];                     // linker col `lane`
        if (lane < clen) mx = s0;
#pragma unroll 4
        for (int ch = 1; ch < 33; ++ch)                // coref block: always visible
          mx = fmaxf(mx, rb[ch * 32 + lane]);
      }
#pragma unroll
      for (int off = 16; off > 0; off >>= 1)
        mx = fmaxf(mx, __shfl_xor(mx, off, 32));

      // ---- pass B: exp-sums for denominator / targets, and target count ----
      float denom = 0.0f, numer = 0.0f;
      int tcnt = 0;
      {
        const float s0 = rb[lane];
        const int lt = link_targets[rowbase * C_ + lane];
        if (lane < clen) {
          const float e = __expf(s0 - mx);
          denom += e;
          if (lt != 0) { numer += e; tcnt++; }
        }
        const int* cidrow = cluster_ids + b * M_;
#pragma unroll 4
        for (int ch = 0; ch < 32; ++ch) {
          const int j = ch * 32 + lane;
          const float e = __expf(rb[C_ + j] - mx);
          denom += e;
          const int cj = cidrow[j];
          const bool tgt = (cidm >= 0) & (cj == cidm) & (j != m);
          if (tgt) { numer += e; tcnt++; }
        }
      }
#pragma unroll
      for (int off = 16; off > 0; off >>= 1) {
        denom += __shfl_xor(denom, off, 32);
        numer += __shfl_xor(numer, off, 32);
        tcnt  += __shfl_xor(tcnt,  off, 32);
      }
      if (tcnt == 0)                                   // singleton -> self-link
        numer += __expf(rb[C_ + m] - mx);

      acc += __logf(denom) - __logf(numer);
    }
  }

  if (lane == 0) wacc[wib] = acc;
  __syncthreads();
  if (threadIdx.x == 0) {
    float s = 0.0f;
#pragma unroll
    for (int w = 0; w < WAVES_PER_BLK; ++w) s += wacc[w];
    partial[blockIdx.x] = s;
  }
}

// Deterministic single-block finalize: fold 512 partials to 256 in LDS, then
// reduce the 256 with V_WMMA_F32_16X16X4_F32 against a ones B-matrix:
// 4 accumulating WMMAs give D[m,n] = sum_c sum_k A_c[m,k]; an 8-component add
// plus one shfl_xor(16) collapses the 16x16 D tile to the scalar total.
__global__ __launch_bounds__(256) void coref_finalize_kernel(
    const float* __restrict__ partial, float* __restrict__ out)
{
  __shared__ float red[256];
  const int t = threadIdx.x;
  red[t] = partial[t] + partial[t + 256];
  __syncthreads();

#if __has_builtin(__builtin_amdgcn_wmma_f32_16x16x4_f32)
  const int wib = t >> 5;
  if (wib == 0) {
    const int L = t & 31;
    const int mrow  = L & 15;
    const int khalf = (L >> 4) << 1;                   // 0 or 2
    v2f ones; ones[0] = 1.0f; ones[1] = 1.0f;          // B = 4x16 all-ones
    v8f d = {};
#pragma unroll
    for (int c = 0; c < 4; ++c) {
      v2f a;                                           // A = 16x4 chunk of partials
      a[0] = red[c * 64 + mrow * 4 + khalf];
      a[1] = red[c * 64 + mrow * 4 + khalf + 1];
      d = __builtin_amdgcn_wmma_f32_16x16x4_f32(
          false, a, false, ones, (short)0, d, false, false);
    }
    float s = ((d[0] + d[1]) + (d[2] + d[3])) + ((d[4] + d[5]) + (d[6] + d[7]));
    s += __shfl_xor(s, 16, 32);                        // add the M=8..15 half
    if (L == 0) out[0] = s;                            // WEIGHT == 1.0
  }
#else
  for (int s = 128; s > 0; s >>= 1) {
    if (t < s) red[t] += red[t + s];
    __syncthreads();
  }
  if (t == 0) out[0] = red[0];
#endif
}

extern "C" void kernel_launch(void* const* d_in, const int* in_sizes, int n_in,
                              void* d_out, int out_size, void* d_ws, size_t ws_size,
                              hipStream_t stream) {
  (void)in_sizes; (void)n_in; (void)out_size; (void)ws_size;
  const float* scores       = (const float*)d_in[0];
  const int*   link_targets = (const int*)d_in[1];
  const int*   cand_len     = (const int*)d_in[2];
  const int*   cluster_ids  = (const int*)d_in[3];
  const int*   lengths      = (const int*)d_in[4];
  float* partial = (float*)d_ws;                 // 512 floats of scratch

  coref_linker_loss_kernel<<<BLOCKS_, THREADS_, 0, stream>>>(
      scores, link_targets, cand_len, cluster_ids, lengths, partial);
  coref_finalize_kernel<<<1, 256, 0, stream>>>(partial, (float*)d_out);
}